// GCN_7172595384884
// MI455X (gfx1250) — compile-verified
//
#include <hip/hip_runtime.h>

// ---------------------------------------------------------------------------
// EdgeConv GNN for MI455X (gfx1250, wave32, WMMA).
// Restructure: out[i] = (deg[i]+1)*(ai[i]+b) + aj[i]
//                       + sum_{e: dst=i} ( aj[src_e] + ea_e @ We.T )
// Dense GEMMs (ai/aj) use V_WMMA_F32_16X16X4_F32 (fp32 exact vs reference).
// Edge scatter uses coalesced global_atomic_add_f32.
// ---------------------------------------------------------------------------

typedef float v2f __attribute__((ext_vector_type(2)));
typedef float v8f __attribute__((ext_vector_type(8)));

#define NNODES 50000
#define NEDGES_ 800000

// ---------------- degree ----------------------------------------------------
__global__ void zero_deg_k(int* deg) {
    int i = blockIdx.x * blockDim.x + threadIdx.x;
    if (i < NNODES) deg[i] = 0;
}

__global__ void count_deg_k(const int* __restrict__ dst, int* __restrict__ deg) {
    int e = blockIdx.x * blockDim.x + threadIdx.x;
    if (e < NEDGES_) atomicAdd(&deg[dst[e]], 1);
}

// ---------------- paired GEMM: ai = H@Wi.T, aj = H@Wj.T ---------------------
// H: NNODES x F (row major).  W: dout x (2F+3) (row major); Wi = cols [0,F),
// Wj = cols [F,2F).  One wave computes one 16x16 tile of both outputs.
template <int F>
__global__ void gemm_pair_k(const float* __restrict__ H,
                            const float* __restrict__ W,
                            float* __restrict__ ai,
                            float* __restrict__ aj,
                            int dout)
{
    const int ldw      = 2 * F + 3;
    const int outTiles = dout >> 4;
    const int lane     = threadIdx.x & 31;
    const int waveId   = threadIdx.x >> 5;
    const int w        = blockIdx.x * (blockDim.x >> 5) + waveId;
    const int total    = (NNODES / 16) * outTiles;
    if (w >= total) return;                       // wave-uniform: EXEC stays full

    const int nodeTile = w / outTiles;
    const int outTile  = w % outTiles;
    const int row   = nodeTile * 16 + (lane & 15);   // A-fragment row (node)
    const int ocol  = outTile  * 16 + (lane & 15);   // B-fragment col (out dim)
    const int khalf = (lane >> 4) * 2;               // K split across lane halves

    v8f ci = {}; v8f cj = {};
    constexpr int KSTEPS = (F + 3) / 4;

    const float* __restrict__ hrow = H + (size_t)row * F;
    const float* __restrict__ wrow = W + (size_t)ocol * ldw;

    #pragma unroll
    for (int ks = 0; ks < KSTEPS; ++ks) {
        const int k0 = ks * 4 + khalf;
        v2f a, bi, bj;
        a.x  = (k0     < F) ? hrow[k0]         : 0.0f;
        a.y  = (k0 + 1 < F) ? hrow[k0 + 1]     : 0.0f;
        bi.x = (k0     < F) ? wrow[k0]         : 0.0f;   // Wi cols [0,F)
        bi.y = (k0 + 1 < F) ? wrow[k0 + 1]     : 0.0f;
        bj.x = (k0     < F) ? wrow[F + k0]     : 0.0f;   // Wj cols [F,2F)
        bj.y = (k0 + 1 < F) ? wrow[F + k0 + 1] : 0.0f;

        ci = __builtin_amdgcn_wmma_f32_16x16x4_f32(false, a, false, bi,
                                                   (short)0, ci, false, false);
        cj = __builtin_amdgcn_wmma_f32_16x16x4_f32(false, a, false, bj,
                                                   (short)0, cj, false, false);
    }

    // C/D layout: VGPR r, lane l -> M = r + (l>=16)*8, N = l&15
    const int mbase = nodeTile * 16 + (lane >> 4) * 8;
    const int n     = outTile  * 16 + (lane & 15);
    #pragma unroll
    for (int r = 0; r < 8; ++r) {
        ai[(size_t)(mbase + r) * dout + n] = ci[r];
        aj[(size_t)(mbase + r) * dout + n] = cj[r];
    }
}

// ---------------- per-node init: out = (deg+1)*(ai+b) + aj ------------------
__global__ void node_init_k(const float* __restrict__ ai,
                            const float* __restrict__ aj,
                            const float* __restrict__ b,
                            const int*   __restrict__ deg,
                            float* __restrict__ out, int dout)
{
    int i = blockIdx.x * blockDim.x + threadIdx.x;
    int total = NNODES * dout;
    if (i >= total) return;
    int node = i / dout;
    int d    = i - node * dout;
    float s = (float)(deg[node] + 1);
    out[i] = s * (ai[i] + b[d]) + aj[i];
}

// ---------------- edge scatter: out[dst] += aj[src] + ea @ We.T -------------
// blockDim.x == dout, one block per edge; lanes coalesce over feature dim.
__global__ void edge_msg_k(const int*   __restrict__ src,
                           const int*   __restrict__ dst,
                           const float* __restrict__ ea,
                           const float* __restrict__ W,   // dout x ldw
                           int ldw, int foff,             // foff = 2F
                           const float* __restrict__ aj,
                           float* __restrict__ out, int dout)
{
    const int e = blockIdx.x;
    const int d = threadIdx.x;
    const int s = src[e];
    const int t = dst[e];
    const float e0 = ea[e * 3 + 0];
    const float e1 = ea[e * 3 + 1];
    const float e2 = ea[e * 3 + 2];
    const float* wr = W + (size_t)d * ldw + foff;          // We row d (L2-resident)
    float v = aj[(size_t)s * dout + d]
            + e0 * wr[0] + e1 * wr[1] + e2 * wr[2];
    unsafeAtomicAdd(&out[(size_t)t * dout + d], v);        // global_atomic_add_f32
}

// ---------------- elementwise ReLU ------------------------------------------
__global__ void relu_k(float* __restrict__ h, int total) {
    int i = blockIdx.x * blockDim.x + threadIdx.x;
    if (i < total) h[i] = fmaxf(h[i], 0.0f);
}

// ---------------- classifier + log_softmax ----------------------------------
__global__ void classifier_k(const float* __restrict__ h,   // N x 32
                             const float* __restrict__ Wc,  // 4 x 32
                             const float* __restrict__ bc,  // 4
                             float* __restrict__ out)       // N x 4
{
    int i = blockIdx.x * blockDim.x + threadIdx.x;
    if (i >= NNODES) return;
    const float* hr = h + (size_t)i * 32;
    float l[4];
    #pragma unroll
    for (int c = 0; c < 4; ++c) {
        float acc = bc[c];
        #pragma unroll
        for (int k = 0; k < 32; ++k) acc += hr[k] * Wc[c * 32 + k];
        l[c] = acc;
    }
    float m = fmaxf(fmaxf(l[0], l[1]), fmaxf(l[2], l[3]));
    float ssum = 0.0f;
    #pragma unroll
    for (int c = 0; c < 4; ++c) ssum += expf(l[c] - m);
    float lse = logf(ssum);
    #pragma unroll
    for (int c = 0; c < 4; ++c) out[(size_t)i * 4 + c] = l[c] - m - lse;
}

// ---------------------------------------------------------------------------
extern "C" void kernel_launch(void* const* d_in, const int* in_sizes, int n_in,
                              void* d_out, int out_size, void* d_ws, size_t ws_size,
                              hipStream_t stream)
{
    const float* x   = (const float*)d_in[0];
    const int*   ei  = (const int*)  d_in[1];
    const float* ea  = (const float*)d_in[2];
    const float* W1  = (const float*)d_in[3];
    const float* b1  = (const float*)d_in[4];
    const float* W2  = (const float*)d_in[5];
    const float* b2  = (const float*)d_in[6];
    const float* W3  = (const float*)d_in[7];
    const float* b3  = (const float*)d_in[8];
    const float* Wc  = (const float*)d_in[9];
    const float* bc  = (const float*)d_in[10];
    float*       out = (float*)d_out;

    const int* src = ei;
    const int* dst = ei + NEDGES_;

    // workspace carve-up (~96 MB total)
    char*  ws  = (char*)d_ws;
    size_t off = 0;
    auto carve = [&](size_t bytes) -> void* {
        void* p = ws + off;
        off += (bytes + 255) & ~(size_t)255;
        return p;
    };
    int*   deg = (int*)  carve((size_t)NNODES * 4);
    float* ai  = (float*)carve((size_t)NNODES * 128 * 4);
    float* aj  = (float*)carve((size_t)NNODES * 128 * 4);
    float* h1  = (float*)carve((size_t)NNODES * 128 * 4);
    float* h2  = (float*)carve((size_t)NNODES * 64 * 4);
    float* h3  = (float*)carve((size_t)NNODES * 32 * 4);

    // in-degree (real edges only; self-loop folded as deg+1)
    zero_deg_k <<<(NNODES + 255) / 256, 256, 0, stream>>>(deg);
    count_deg_k<<<(NEDGES_ + 255) / 256, 256, 0, stream>>>(dst, deg);

    // ---- layer 1: F=3 -> 128, ReLU ----
    {
        const int dout = 128;
        const int tiles = (NNODES / 16) * (dout / 16);
        gemm_pair_k<3><<<(tiles + 3) / 4, 128, 0, stream>>>(x, W1, ai, aj, dout);
        node_init_k<<<(NNODES * dout + 255) / 256, 256, 0, stream>>>(ai, aj, b1, deg, h1, dout);
        edge_msg_k<<<NEDGES_, dout, 0, stream>>>(src, dst, ea, W1, 2 * 3 + 3, 2 * 3, aj, h1, dout);
        relu_k<<<(NNODES * dout + 255) / 256, 256, 0, stream>>>(h1, NNODES * dout);
    }
    // ---- layer 2: F=128 -> 64, ReLU ----
    {
        const int dout = 64;
        const int tiles = (NNODES / 16) * (dout / 16);
        gemm_pair_k<128><<<(tiles + 3) / 4, 128, 0, stream>>>(h1, W2, ai, aj, dout);
        node_init_k<<<(NNODES * dout + 255) / 256, 256, 0, stream>>>(ai, aj, b2, deg, h2, dout);
        edge_msg_k<<<NEDGES_, dout, 0, stream>>>(src, dst, ea, W2, 2 * 128 + 3, 2 * 128, aj, h2, dout);
        relu_k<<<(NNODES * dout + 255) / 256, 256, 0, stream>>>(h2, NNODES * dout);
    }
    // ---- layer 3: F=64 -> 32, no ReLU ----
    {
        const int dout = 32;
        const int tiles = (NNODES / 16) * (dout / 16);
        gemm_pair_k<64><<<(tiles + 3) / 4, 128, 0, stream>>>(h2, W3, ai, aj, dout);
        node_init_k<<<(NNODES * dout + 255) / 256, 256, 0, stream>>>(ai, aj, b3, deg, h3, dout);
        edge_msg_k<<<NEDGES_, dout, 0, stream>>>(src, dst, ea, W3, 2 * 64 + 3, 2 * 64, aj, h3, dout);
    }
    // ---- classifier + log_softmax ----
    classifier_k<<<(NNODES + 255) / 256, 256, 0, stream>>>(h3, Wc, bc, out);
}